// GIN_Custom_31499290149084
// MI455X (gfx1250) — compile-verified
//
#include <hip/hip_runtime.h>
#include <cmath>

#define N_NODES 50000
#define N_EDGES 800000
#define DF 128

typedef __attribute__((ext_vector_type(2))) float v2f;
typedef __attribute__((ext_vector_type(8))) float v8f;

// ---------------------------------------------------------------------------
// h0 = x   (init aggregation buffer with (1+eps)*x, eps = 0)
// ---------------------------------------------------------------------------
__global__ void copy_kernel(float* __restrict__ dst, const float* __restrict__ src, int n4) {
    int i = blockIdx.x * blockDim.x + threadIdx.x;
    if (i < n4) ((float4*)dst)[i] = ((const float4*)src)[i];
}

// ---------------------------------------------------------------------------
// h0[dst] += x[src]  for every edge.  32 lanes per edge, float4 per lane.
// ---------------------------------------------------------------------------
__global__ void scatter_kernel(float* __restrict__ agg, const float* __restrict__ x,
                               const int* __restrict__ ei) {
    int t = blockIdx.x * blockDim.x + threadIdx.x;
    int e = t >> 5;
    if (e >= N_EDGES) return;
    int f = (t & 31) << 2;
    int s = ei[e];
    int d = ei[N_EDGES + e];
    float4 v = *(const float4*)(x + (size_t)s * DF + f);
    float* p = agg + (size_t)d * DF + f;
    atomicAdd(p + 0, v.x);
    atomicAdd(p + 1, v.y);
    atomicAdd(p + 2, v.z);
    atomicAdd(p + 3, v.w);
}

// ---------------------------------------------------------------------------
// WMMA f32 GEMM: out[M x 128] = epilogue( sum_seg A_seg[M x 128] @ W_seg[128 x 128] + bias )
//   EPI 0: + bias                      (JK head)
//   EPI 1: relu(+ bias)                (MLP hidden)
//   EPI 2: elu(+ bias + residual)      (MLP out + skip)
// One block = 16 rows x 128 cols; 8 waves, each owns a 16x16 tile.
// A tile staged via LDS; W read from global (L2-resident, shared by all blocks).
// ---------------------------------------------------------------------------
template<int NSEG, int EPI>
__global__ void __launch_bounds__(256, 2)
gemm_wmma(const float* __restrict__ A0, const float* __restrict__ A1,
          const float* __restrict__ A2, const float* __restrict__ W,
          const float* __restrict__ bias, const float* __restrict__ resid,
          float* __restrict__ out) {
    __shared__ float lA[16 * DF];

    const int lane = threadIdx.x & 31;
    const int wave = threadIdx.x >> 5;
    const int r    = lane & 15;   // row (A) / col (B,C)
    const int hi   = lane >> 4;   // half-wave select
    const int n0   = wave * 16;   // this wave's output-column tile
    const int row0 = blockIdx.x * 16;

    v8f acc = {0.f, 0.f, 0.f, 0.f, 0.f, 0.f, 0.f, 0.f};

    const float* As[3] = {A0, A1, A2};
    #pragma unroll
    for (int seg = 0; seg < NSEG; ++seg) {
        // Cooperative stage of the 16x128 f32 A tile (8 KB) into LDS.
        const float4* s4 = (const float4*)(As[seg] + (size_t)row0 * DF);
        float4* d4 = (float4*)lA;
        #pragma unroll
        for (int i = 0; i < 2; ++i)
            d4[threadIdx.x + i * 256] = s4[threadIdx.x + i * 256];
        __syncthreads();

        const float* Wseg = W + seg * DF * DF;
        #pragma unroll
        for (int k0 = 0; k0 < DF; k0 += 4) {
            // A 16x4 fragment: lane r holds A[r][k0+2*hi], A[r][k0+2*hi+1]
            const float* ap = lA + r * DF + k0 + 2 * hi;
            v2f a; a.x = ap[0]; a.y = ap[1];
            // B 4x16 fragment: lane holds W[k0+2*hi][n0+r], W[k0+2*hi+1][n0+r]
            const float* bp = Wseg + (k0 + 2 * hi) * DF + n0 + r;
            v2f b; b.x = bp[0]; b.y = bp[DF];
            acc = __builtin_amdgcn_wmma_f32_16x16x4_f32(
                false, a, false, b, (short)0, acc, false, false);
        }
        __syncthreads();
    }

    const int col = n0 + r;
    const float bb = bias[col];
    #pragma unroll
    for (int v = 0; v < 8; ++v) {
        int row = row0 + v + 8 * hi;            // C/D layout: VGPR v -> M = v + 8*hi
        float y = acc[v] + bb;
        if (EPI == 1) y = fmaxf(y, 0.0f);
        if (EPI == 2) {
            y += resid[(size_t)row * DF + col];
            y = (y > 0.0f) ? y : expm1f(y);      // elu, alpha = 1
        }
        out[(size_t)row * DF + col] = y;
    }
}

// ---------------------------------------------------------------------------
extern "C" void kernel_launch(void* const* d_in, const int* in_sizes, int n_in,
                              void* d_out, int out_size, void* d_ws, size_t ws_size,
                              hipStream_t stream) {
    const float* x   = (const float*)d_in[0];
    const int*   ei  = (const int*)d_in[1];
    const float* W1[3] = {(const float*)d_in[2],  (const float*)d_in[6],  (const float*)d_in[10]};
    const float* b1[3] = {(const float*)d_in[3],  (const float*)d_in[7],  (const float*)d_in[11]};
    const float* W2[3] = {(const float*)d_in[4],  (const float*)d_in[8],  (const float*)d_in[12]};
    const float* b2[3] = {(const float*)d_in[5],  (const float*)d_in[9],  (const float*)d_in[13]};
    const float* Wjk = (const float*)d_in[14];
    const float* bjk = (const float*)d_in[15];

    const size_t NM = (size_t)N_NODES * DF;
    float* h0 = (float*)d_ws;                       // agg buffer, reused as relu intermediate
    float* xs[3] = {h0 + NM, h0 + 2 * NM, h0 + 3 * NM};  // per-layer outputs for JK

    const int n4 = (int)(NM / 4);
    dim3 blk(256);
    const int gBlocks = N_NODES / 16;               // 50000 = 3125 * 16, exact

    const float* xin = x;
    for (int l = 0; l < 3; ++l) {
        copy_kernel<<<(n4 + 255) / 256, blk, 0, stream>>>(h0, xin, n4);
        scatter_kernel<<<(N_EDGES * 32) / 256, blk, 0, stream>>>(h0, xin, ei);
        // h0 = relu(h0 @ W1 + b1)   (in-place safe: each block reads its own tile
        //                            into LDS before writing the same rows back)
        gemm_wmma<1, 1><<<gBlocks, blk, 0, stream>>>(h0, nullptr, nullptr,
                                                     W1[l], b1[l], nullptr, h0);
        // xs[l] = elu(h0 @ W2 + b2 + xin)
        gemm_wmma<1, 2><<<gBlocks, blk, 0, stream>>>(h0, nullptr, nullptr,
                                                     W2[l], b2[l], xin, xs[l]);
        xin = xs[l];
    }
    // out = concat(xs) @ Wjk + bjk  ==  sum_l xs[l] @ Wjk[l*128:(l+1)*128] + bjk
    gemm_wmma<3, 0><<<gBlocks, blk, 0, stream>>>(xs[0], xs[1], xs[2],
                                                 Wjk, bjk, nullptr, (float*)d_out);
}